// SegmentAwareLossModuleV3_48318382080474
// MI455X (gfx1250) — compile-verified
//
#include <hip/hip_runtime.h>
#include <hip/hip_bf16.h>

typedef float v2f __attribute__((ext_vector_type(2)));
typedef float v8f __attribute__((ext_vector_type(8)));

#define EPS2      (1e-3f * 1e-3f)
#define ALPHA_C   0.45f
#define PIX       (512 * 512)

// (sq + eps^2)^0.45 ; sq >= 0 so log argument is always positive
__device__ __forceinline__ float charb_sq(float sq) {
    float v = sq + EPS2;
    return __builtin_exp2f(ALPHA_C * __builtin_log2f(v));
}

__device__ __forceinline__ float wave_sum(float v) {
    #pragma unroll
    for (int o = 16; o > 0; o >>= 1) v += __shfl_xor(v, o, 32);
    return v;
}

// ---------------------------------------------------------------------------
// Workspace layout (floats):
//   [0    .. 1023] moments  : b*256 + s*16 + n   (n = 0..14 used)
//   [1024 .. 2047] derived  : b*256 + s*16 + k   (k: au,bu,cu,av,bv,cv,mu,mv,valid,ms)
//   [2048 .. 2063] mask_sum_s[16]   (raw sum of m over b,p)
//   [2064 .. 2079] denomsum[16]     (sum over b of clipped ms)
//   [2080 .. 2095] hnum[16]         (homography numerator per segment)
//   [2096 .. 2111] vnum[16]         (motion variance numerator per segment)
//   [2112] bsum, [2113] dsum
// ---------------------------------------------------------------------------

__global__ void k0_zero(float* __restrict__ ws) {
    for (int i = threadIdx.x; i < 4096; i += blockDim.x) ws[i] = 0.0f;
}

// ---------------------------------------------------------------------------
// Kernel 1: moment reduction with V_WMMA_F32_16X16X4_F32.
// D[s, n] += sum_k A[s, k] * B[k, n] over 4-pixel chunks.
//   WMMA #1: A = (m > 0.5),  B cols 0..11 = {xx,xy,x, yy,y,1, x*fu,y*fu,fu, x*fv,y*fv,fv}
//   WMMA #2: A = m,          B cols 12..14 = {1, fu, fv}   (same accumulator)
// B entries built branchlessly from per-lane loop-invariant 0/1 coefficients.
// Per-wave partials are reduced across the block in LDS (ds_add_f32), then a
// single global atomic per value per block (8x less L2 atomic serialization).
// grid: (32, 4) x 256 threads (8 waves). Each wave: 256 chunks of 4 pixels.
// ---------------------------------------------------------------------------
__global__ void __launch_bounds__(256)
k1_moments(const float* __restrict__ flow,
           const float* __restrict__ masks,
           float* __restrict__ ws) {
    const int b     = blockIdx.y;
    const int wave  = threadIdx.x >> 5;
    const int lane  = threadIdx.x & 31;
    const int n     = lane & 15;     // segment row for A, moment column for B
    const int khalf = lane >> 4;     // which K-pair this lane holds (K0/K1 vs K2/K3)

    __shared__ float mred[256];
    mred[threadIdx.x] = 0.0f;
    __syncthreads();

    const float* fu_p   = flow  + (size_t)b * 2 * PIX;
    const float* fv_p   = fu_p  + PIX;
    const float* mplane = masks + ((size_t)b * 16 + n) * PIX;

    // Per-lane 0/1 coefficients: moment_n = P1 * P2
    //   P1 = cx*x + cy*y + c1        P2 = dx*x + dy*y + dfu*fu + dfv*fv + d1
    //   b2 = e1 + efu*fu + efv*fv    (m-weighted columns 12..14)
    const float cx  = (n == 0 || n == 1 || n == 2 || n == 6 || n == 9) ? 1.0f : 0.0f;
    const float cy  = (n == 3 || n == 4 || n == 7 || n == 10)          ? 1.0f : 0.0f;
    const float c1  = (n == 5 || n == 8 || n == 11)                    ? 1.0f : 0.0f;
    const float dx  = (n == 0)                                         ? 1.0f : 0.0f;
    const float dy  = (n == 1 || n == 3)                               ? 1.0f : 0.0f;
    const float dfu = (n >= 6 && n <= 8)                               ? 1.0f : 0.0f;
    const float dfv = (n >= 9 && n <= 11)                              ? 1.0f : 0.0f;
    const float d1  = (n == 2 || n == 4 || n == 5)                     ? 1.0f : 0.0f;
    const float e1  = (n == 12)                                        ? 1.0f : 0.0f;
    const float efu = (n == 13)                                        ? 1.0f : 0.0f;
    const float efv = (n == 14)                                        ? 1.0f : 0.0f;

    v8f acc = {};
    const int chunk0 = blockIdx.x * 2048 + wave * 256;

    for (int i = 0; i < 256; ++i) {
        const int p = (chunk0 + i) * 4 + 2 * khalf;   // even -> 8B aligned
        // prefetch the mask stream ~2KB ahead (HBM-bound first pass)
        __builtin_prefetch(mplane + p + 512, 0, 1);

        // chunk never crosses a row (512 % 4 == 0): y shared by both pixels
        const int   col = p & 511;
        const int   row = p >> 9;
        const float y   = (float)row * (2.0f / 511.0f) - 1.0f;
        const float xA  = (float)col * (2.0f / 511.0f) - 1.0f;

        const v2f mv  = *(const v2f*)(mplane + p);
        const v2f fuv = *(const v2f*)(fu_p   + p);
        const v2f fvv = *(const v2f*)(fv_p   + p);

        v2f A1, A2, B1, B2;
        #pragma unroll
        for (int e = 0; e < 2; ++e) {
            const float x  = xA + (float)e * (2.0f / 511.0f);
            const float fu = fuv[e];
            const float fv = fvv[e];
            const float m  = mv[e];
            const float P1 = fmaf(cx, x, fmaf(cy, y, c1));
            const float P2 = fmaf(dx, x, fmaf(dy, y, fmaf(dfu, fu, fmaf(dfv, fv, d1))));
            A1[e] = (m > 0.5f) ? 1.0f : 0.0f;
            A2[e] = m;
            B1[e] = P1 * P2;
            B2[e] = fmaf(efu, fu, fmaf(efv, fv, e1));
        }
        acc = __builtin_amdgcn_wmma_f32_16x16x4_f32(false, A1, false, B1,
                                                    (short)0, acc, false, false);
        acc = __builtin_amdgcn_wmma_f32_16x16x4_f32(false, A2, false, B2,
                                                    (short)0, acc, false, false);
    }

    // C/D layout: VGPR j, lanes 0-15: (M=j, N=lane); lanes 16-31: (M=j+8, N=lane-16)
    #pragma unroll
    for (int j = 0; j < 8; ++j) {
        const int s = j + 8 * khalf;
        atomicAdd(&mred[s * 16 + n], acc[j]);   // LDS atomic, <=8-way contention
    }
    __syncthreads();
    float* mom = ws + b * 256;
    atomicAdd(&mom[threadIdx.x], mred[threadIdx.x]);  // 1 global atomic/value/block
}

// ---------------------------------------------------------------------------
// Kernel 2: per-(b,s) 3x3 symmetric solve (Cramer/adjugate), 64 threads.
// ---------------------------------------------------------------------------
__global__ void k2_solve(float* __restrict__ ws) {
    const int t = threadIdx.x;
    if (t >= 64) return;
    const int b = t >> 4, s = t & 15;
    const float* M = ws + b * 256 + s * 16;
    const float Sxx = M[0],  Sxy = M[1],  Sx  = M[2];
    const float Syy = M[3],  Sy  = M[4],  Sw  = M[5];
    const float Xu  = M[6],  Yu  = M[7],  Cu  = M[8];
    const float Xv  = M[9],  Yv  = M[10], Cv  = M[11];
    const float Sm  = M[12], Su  = M[13], Sv  = M[14];

    const float A00 = Sxx + 1e-6f, A01 = Sxy, A02 = Sx;
    const float A11 = Syy + 1e-6f, A12 = Sy;
    const float A22 = Sw  + 1e-6f;

    const float c00 = A11 * A22 - A12 * A12;
    const float c01 = A02 * A12 - A01 * A22;
    const float c02 = A01 * A12 - A02 * A11;
    const float c11 = A00 * A22 - A02 * A02;
    const float c12 = A01 * A02 - A00 * A12;
    const float c22 = A00 * A11 - A01 * A01;
    const float det = A00 * c00 + A01 * c01 + A02 * c02;
    const float inv = 1.0f / det;

    float* D = ws + 1024 + b * 256 + s * 16;
    D[0] = (c00 * Xu + c01 * Yu + c02 * Cu) * inv;
    D[1] = (c01 * Xu + c11 * Yu + c12 * Cu) * inv;
    D[2] = (c02 * Xu + c12 * Yu + c22 * Cu) * inv;
    D[3] = (c00 * Xv + c01 * Yv + c02 * Cv) * inv;
    D[4] = (c01 * Xv + c11 * Yv + c12 * Cv) * inv;
    D[5] = (c02 * Xv + c12 * Yv + c22 * Cv) * inv;

    const float ms = fmaxf(Sm, 1e-6f);
    D[6] = Su / ms;
    D[7] = Sv / ms;
    D[8] = (Sw >= 100.0f) ? 1.0f : 0.0f;   // n_valid >= MIN_SEG_PIX
    D[9] = ms;

    atomicAdd(&ws[2048 + s], Sm);  // raw mask_sum_s
    atomicAdd(&ws[2064 + s], ms);  // sum of clipped ms
}

// ---------------------------------------------------------------------------
// Kernel 3: fused per-pixel pass: homography residual loss, motion variance,
// boundary + discontinuity losses. grid (64, 4) x 256 threads.
// Per-wave sums -> LDS atomics -> one global atomic per value per block.
// ---------------------------------------------------------------------------
__global__ void __launch_bounds__(256)
k3_pixel(const float* __restrict__ flow,
         const float* __restrict__ masks,
         const float* __restrict__ bmap,
         float* __restrict__ ws) {
    const int b = blockIdx.y;
    __shared__ float sder[16 * 10];
    __shared__ float red[34];
    if (threadIdx.x < 160) {
        const int s = threadIdx.x / 10, k = threadIdx.x % 10;
        sder[threadIdx.x] = ws[1024 + b * 256 + s * 16 + k];
    }
    if (threadIdx.x < 34) red[threadIdx.x] = 0.0f;
    __syncthreads();

    const float* fu_p = flow + (size_t)b * 2 * PIX;
    const float* fv_p = fu_p + PIX;
    const float* mb   = masks + (size_t)b * 16 * PIX;
    const float* bm   = bmap + (size_t)b * PIX;

    float hnum[16], vnum[16];
    #pragma unroll
    for (int s = 0; s < 16; ++s) { hnum[s] = 0.0f; vnum[s] = 0.0f; }
    float bsum = 0.0f, dsum = 0.0f;

    const int tid = blockIdx.x * blockDim.x + threadIdx.x;  // 0..16383
    for (int p = tid; p < PIX; p += 64 * 256) {
        const int col = p & 511, row = p >> 9;
        const float x  = (float)col * (2.0f / 511.0f) - 1.0f;
        const float y  = (float)row * (2.0f / 511.0f) - 1.0f;
        const float fu = fu_p[p], fv = fv_p[p];

        // flow gradient magnitude: clamp neighbor index at far edge so that
        // the difference is exactly zero there (matches zero-padding), branchless
        const int pr = (col < 511) ? p + 1   : p;
        const int pd = (row < 511) ? p + 512 : p;
        const float dxu = fu_p[pr] - fu, dxv = fv_p[pr] - fv;
        const float dyu = fu_p[pd] - fu, dyv = fv_p[pd] - fv;
        const float fxm = sqrtf(dxu * dxu + dxv * dxv);
        const float fym = sqrtf(dyu * dyu + dyv * dyv);
        const float fgm = 0.5f * (fxm + fym);

        const float bv       = bm[p];
        const float interior = fminf(fmaxf(1.0f - bv, 0.0f), 1.0f);
        bsum += charb_sq(fgm * fgm) * interior;
        dsum += fmaxf(0.5f - fgm, 0.0f) * bv;

        #pragma unroll
        for (int s = 0; s < 16; ++s) {
            const float m = mb[s * PIX + p];          // 32-bit offsets
            const float* d = &sder[s * 10];
            const float valid = d[8];
            const float ru = (fu - fmaf(d[0], x, fmaf(d[1], y, d[2]))) * valid;
            const float rv = (fv - fmaf(d[3], x, fmaf(d[4], y, d[5]))) * valid;
            const float lm = 0.5f * (charb_sq(ru * ru) + charb_sq(rv * rv));
            hnum[s] += lm * m;
            const float du = fu - d[6], dv = fv - d[7];
            vnum[s] += charb_sq(du * du + dv * dv) * m;   // charb(|diff|)
        }
        // pull next tile's mask lines toward the WGP (L2-resident second pass)
        __builtin_prefetch(mb + p + 64 * 256, 0, 1);
    }

    const int lane = threadIdx.x & 31;
    #pragma unroll
    for (int s = 0; s < 16; ++s) {
        float v = wave_sum(hnum[s]);
        if (lane == 0) atomicAdd(&red[s], v);
        v = wave_sum(vnum[s]);
        if (lane == 0) atomicAdd(&red[16 + s], v);
    }
    float v = wave_sum(bsum);
    if (lane == 0) atomicAdd(&red[32], v);
    v = wave_sum(dsum);
    if (lane == 0) atomicAdd(&red[33], v);

    __syncthreads();
    if (threadIdx.x < 34) atomicAdd(&ws[2080 + threadIdx.x], red[threadIdx.x]);
}

// ---------------------------------------------------------------------------
// Kernel 4: combine everything to the scalar loss.
// ---------------------------------------------------------------------------
__global__ void k4_final(const float* __restrict__ ws, float* __restrict__ out) {
    float tl = 0.0f, tw = 0.0f;
    for (int s = 0; s < 16; ++s) {
        const float msum = ws[2048 + s];
        const float keep = (msum >= 400.0f) ? 1.0f : 0.0f;  // MIN_SEG_PIX * B
        tl += ws[2080 + s] * keep;
        tw += msum * keep;
    }
    const float homo = 0.1f * ((tw > 0.0f) ? tl / fmaxf(tw, 1e-6f) : 0.0f);

    float tot = 0.0f, sw = 0.0f;
    for (int s = 0; s < 16; ++s) {
        const float segw = ws[2064 + s];
        const float varc = ws[2096 + s] / fmaxf(segw, 1e-6f);
        tot += varc * segw;
        sw  += segw;
    }
    const float motion = 0.05f * ((sw > 1e-6f) ? tot / sw : 0.0f);

    const float NP = 4.0f * 512.0f * 512.0f;
    const float bl = 0.05f * ws[2112] / NP;
    const float dl = 0.02f * ws[2113] / NP;
    out[0] = homo + motion + bl + dl;
}

extern "C" void kernel_launch(void* const* d_in, const int* in_sizes, int n_in,
                              void* d_out, int out_size, void* d_ws, size_t ws_size,
                              hipStream_t stream) {
    const float* flow  = (const float*)d_in[0];   // [4,2,512,512]
    const float* masks = (const float*)d_in[1];   // [4,16,512,512]
    const float* bmap  = (const float*)d_in[2];   // [4,1,512,512]
    float* ws  = (float*)d_ws;
    float* out = (float*)d_out;

    k0_zero<<<1, 256, 0, stream>>>(ws);
    k1_moments<<<dim3(32, 4), 256, 0, stream>>>(flow, masks, ws);
    k2_solve<<<1, 64, 0, stream>>>(ws);
    k3_pixel<<<dim3(64, 4), 256, 0, stream>>>(flow, masks, bmap, ws);
    k4_final<<<1, 1, 0, stream>>>(ws, out);
}